// GCNConv_9216999817888
// MI455X (gfx1250) — compile-verified
//
#include <hip/hip_runtime.h>

#define BATCH   4
#define NNODES  50000
#define NEDGES  800000
#define FIN     128
#define FOUT    128
#define MROWS   (BATCH * NNODES)   // 200000 rows total, divisible by 64

typedef float v2f __attribute__((ext_vector_type(2)));
typedef float v8f __attribute__((ext_vector_type(8)));

// ---------------------------------------------------------------------------
// Kernel 1: h = x @ W^T via V_WMMA_F32_16X16X4_F32 (fp32, K=4 per step).
//
// Block = 256 threads = 8 waves. Wave w computes output-feature tile
// col0 = 16*w. Block covers a 64-row M-block (4 sub-tiles of 16 rows),
// so each wave keeps 4 accumulator tiles and reuses its B fragment 4x.
//
// Fragment layouts (CDNA5 ISA 7.12.2, wave32):
//   A (16x4):  element (m = lane%16, k = v + 2*(lane/16)), v = VGPR 0..1
//   B (4x16):  element (k = v + 2*(lane/16), n = lane%16)
//   C/D(16x16):element (m = v + 8*(lane/16), n = lane%16), v = VGPR 0..7
// Both A and B per-lane elements are K-consecutive pairs -> float2 loads.
// ---------------------------------------------------------------------------
__global__ void __launch_bounds__(256)
gcn_gemm_wmma(const float* __restrict__ x,   // [MROWS, FIN] row-major
              const float* __restrict__ W,   // [FOUT, FIN]  row-major
              float* __restrict__ h)         // [MROWS, FOUT]
{
    const int lane = threadIdx.x & 31;
    const int wave = threadIdx.x >> 5;    // 0..7 -> feature tile
    const int idx  = lane & 15;           // m (for A) / n (for B) within tile
    const int half = lane >> 4;           // 0 or 1 -> K offset 0 or 2
    const int col0 = wave * 16;
    const int rowBase = blockIdx.x * 64;

    v8f acc0 = {}; v8f acc1 = {}; v8f acc2 = {}; v8f acc3 = {};

    // B[k][n] = W[col0 + n][k]  -> gather along W row (col0+idx)
    const float* wp = W + (size_t)(col0 + idx) * FIN + 2 * half;
    const float* xp = x + (size_t)(rowBase + idx) * FIN + 2 * half;

#pragma unroll 4
    for (int k0 = 0; k0 < FIN; k0 += 4) {
        v2f bf = *(const v2f*)(wp + k0);
        v2f a0 = *(const v2f*)(xp + k0);
        v2f a1 = *(const v2f*)(xp + 16 * FIN + k0);
        v2f a2 = *(const v2f*)(xp + 32 * FIN + k0);
        v2f a3 = *(const v2f*)(xp + 48 * FIN + k0);
        // args: (neg_a, A, neg_b, B, c_mod, C, reuse_a, reuse_b)
        acc0 = __builtin_amdgcn_wmma_f32_16x16x4_f32(false, a0, false, bf, (short)0, acc0, false, false);
        acc1 = __builtin_amdgcn_wmma_f32_16x16x4_f32(false, a1, false, bf, (short)0, acc1, false, false);
        acc2 = __builtin_amdgcn_wmma_f32_16x16x4_f32(false, a2, false, bf, (short)0, acc2, false, false);
        acc3 = __builtin_amdgcn_wmma_f32_16x16x4_f32(false, a3, false, bf, (short)0, acc3, false, false);
    }

    float* hp = h + (size_t)rowBase * FOUT + col0 + idx;
#pragma unroll
    for (int v = 0; v < 8; ++v) {
        const int m = v + 8 * half;
        hp[(size_t)(m)      * FOUT] = acc0[v];
        hp[(size_t)(m + 16) * FOUT] = acc1[v];
        hp[(size_t)(m + 32) * FOUT] = acc2[v];
        hp[(size_t)(m + 48) * FOUT] = acc3[v];
    }
}

// ---------------------------------------------------------------------------
// Kernel 2: zero the output (harness poisons d_out with 0xAA).
// out has MROWS*FOUT = 25,600,000 floats = 6,400,000 float4.
// ---------------------------------------------------------------------------
__global__ void __launch_bounds__(256)
gcn_zero(float4* __restrict__ out)
{
    const size_t i = (size_t)blockIdx.x * 256 + threadIdx.x;
    out[i] = make_float4(0.f, 0.f, 0.f, 0.f);
}

// ---------------------------------------------------------------------------
// Kernel 3: edge-parallel scatter:  out[b,r,:] += val * h[b,c,:]
// One thread per (edge, feature); batch loop inside. h and out slices are
// L2-resident (25.6 MB each vs 192 MB L2), so gathers + fp32 atomics stay
// in-cache; HBM only sees h fill + out writeback.
// ---------------------------------------------------------------------------
__global__ void __launch_bounds__(256)
gcn_spmm_scatter(const float* __restrict__ h,
                 const int*   __restrict__ erow,
                 const int*   __restrict__ ecol,
                 const float* __restrict__ evals,
                 float* __restrict__ out)
{
    const unsigned gid = blockIdx.x * 256u + threadIdx.x;  // < NEDGES*128
    const int e = gid >> 7;
    const int f = gid & 127;
    const int r = erow[e];
    const int c = ecol[e];
    const float v = evals[e];

#pragma unroll
    for (int b = 0; b < BATCH; ++b) {
        const size_t hi = ((size_t)b * NNODES + c) * (size_t)FOUT + f;
        const size_t oi = ((size_t)b * NNODES + r) * (size_t)FOUT + f;
        unsafeAtomicAdd(&out[oi], v * h[hi]);   // -> global_atomic_add_f32
    }
}

// ---------------------------------------------------------------------------
extern "C" void kernel_launch(void* const* d_in, const int* in_sizes, int n_in,
                              void* d_out, int out_size, void* d_ws, size_t ws_size,
                              hipStream_t stream)
{
    const float* x     = (const float*)d_in[0];   // [B,N,FIN]
    const float* W     = (const float*)d_in[1];   // [FOUT,FIN]
    const int*   erow  = (const int*)  d_in[2];   // [E]
    const int*   ecol  = (const int*)  d_in[3];   // [E]
    const float* evals = (const float*)d_in[4];   // [E]
    float* out = (float*)d_out;                   // [B,N,FOUT]
    float* h   = (float*)d_ws;                    // [B*N, FOUT] scratch (102.4 MB)

    // 1) projection: 200000 rows / 64 per block = 3125 blocks
    gcn_gemm_wmma<<<MROWS / 64, 256, 0, stream>>>(x, W, h);

    // 2) zero output: 25.6M floats -> 6.4M float4 / 256 = 25000 blocks
    gcn_zero<<<(MROWS * FOUT) / 4 / 256, 256, 0, stream>>>((float4*)out);

    // 3) scatter: E*128 threads / 256 = 400000 blocks
    gcn_spmm_scatter<<<(NEDGES * 128) / 256, 256, 0, stream>>>(h, erow, ecol, evals, out);
}